// RingMemoryModel_15882789061041
// MI455X (gfx1250) — compile-verified
//
#include <hip/hip_runtime.h>
#include <hip/hip_bf16.h>
#include <math.h>

// Problem constants (from reference)
#define Bsz   256
#define Tlen  1024
#define Idim  64
#define Pdim  256
#define Edim  128
#define NOUT  64
#define TEMP_INV 0.125f   // 1/8
#define LN_EPS 1e-5f

typedef __attribute__((ext_vector_type(2))) float v2f;
typedef __attribute__((ext_vector_type(8))) float v8f;

// Hardware tanh (V_TANH_F32) if this toolchain exposes it; libm fallback.
#if __has_builtin(__builtin_amdgcn_tanhf)
  #define FAST_TANH(x) __builtin_amdgcn_tanhf(x)
#elif __has_builtin(__builtin_amdgcn_tanh_f32)
  #define FAST_TANH(x) __builtin_amdgcn_tanh_f32(x)
#else
  #define FAST_TANH(x) tanhf(x)
#endif

// ---------------------------------------------------------------------------
// Phase 1: EMB[b*T+t][e] = tanh( X[b,t,:] @ W_in[e,:] + b_in[e] )
// Pure fp32 WMMA: V_WMMA_F32_16X16X4_F32, K-loop of 16 over I=64.
// One wave computes one 16x16 tile of the [BT=262144, E=128] output.
// A-fragment layout (ISA 32-bit 16x4 table): lanes 0-15: K={k0, k0+1} in
// (v0,v1); lanes 16-31: K={k0+2, k0+3}. M = lane&15. B mirrored (N = lane&15).
// ---------------------------------------------------------------------------
__global__ __launch_bounds__(256)
void emb_wmma_kernel(const float* __restrict__ X,      // [BT, 64]
                     const float* __restrict__ Win,    // [128, 64]
                     const float* __restrict__ b_in,   // [128]
                     float* __restrict__ EMB)          // [BT, 128]
{
    const int wave = threadIdx.x >> 5;
    const int lane = threadIdx.x & 31;
    const long tile  = (long)blockIdx.x * 8 + wave;    // 131072 tiles total
    const long mtile = tile >> 3;                      // 0..16383
    const int  ntile = (int)(tile & 7);                // 0..7
    const long m0 = mtile * 16;
    const int  n0 = ntile * 16;
    const int  lm = lane & 15;
    const int  kh = lane >> 4;                         // 0 or 1

    const float* __restrict__ arow = X   + (m0 + lm) * Idim;
    const float* __restrict__ brow = Win + (long)(n0 + lm) * Idim;

    v8f c = {};
#pragma unroll
    for (int k0 = 0; k0 < Idim; k0 += 4) {
        v2f a, b;
        a.x = arow[k0 + 2 * kh + 0];
        a.y = arow[k0 + 2 * kh + 1];
        b.x = brow[k0 + 2 * kh + 0];
        b.y = brow[k0 + 2 * kh + 1];
        c = __builtin_amdgcn_wmma_f32_16x16x4_f32(
                /*neg_a=*/false, a, /*neg_b=*/false, b,
                /*c_mod=*/(short)0, c, /*reuse_a=*/false, /*reuse_b=*/false);
    }

    const int col = n0 + lm;
    const float bb = b_in[col];
#pragma unroll
    for (int j = 0; j < 8; ++j) {
        long row = m0 + j + 8 * kh;                    // C layout: VGPRj -> M=j / j+8
        EMB[row * (long)Edim + col] = FAST_TANH(c[j] + bb);
    }
}

// ---------------------------------------------------------------------------
// Wave32 butterfly sum (all lanes end with the total).
// ---------------------------------------------------------------------------
__device__ __forceinline__ float wave_sum32(float v) {
#pragma unroll
    for (int off = 16; off > 0; off >>= 1)
        v += __shfl_xor(v, off, 32);
    return v;
}

// ---------------------------------------------------------------------------
// Phase 2: one wave32 per batch. mem[P=256][E=128] lives entirely in LDS
// (128 KB of the WGP's 320 KB), jump_dest table (1 KB) cached behind it.
// Lane owns E-elements [4*lane, 4*lane+4). Single wave => no barriers; LDS
// is in-order per wave and reductions use 32-wide shfl butterflies.
// ---------------------------------------------------------------------------
__global__ __launch_bounds__(32)
void ring_seq_kernel(const float* __restrict__ x,            // [B,T,64]
                     const float* __restrict__ pointer_init, // [B]
                     const float* __restrict__ Win,          // [128,64]
                     const float* __restrict__ b_in,         // [128]
                     const float* __restrict__ ln_w,         // [128]
                     const float* __restrict__ ln_b,         // [128]
                     const float* __restrict__ jump_dest,    // [256]
                     const float* __restrict__ Wg,           // [1,128]
                     const float* __restrict__ bg,           // [1]
                     const float* __restrict__ ctx_strength, // [1]
                     const float* __restrict__ Wo,           // [64,128]
                     const float* __restrict__ bo,           // [64]
                     const float* __restrict__ EMB,          // [B*T,128] (if precomp)
                     int use_precomp,
                     float* __restrict__ out)                // [B,64]
{
    extern __shared__ float lds[];
    float* mem = lds;                 // [P*E]  ring memory, 128 KB
    float* jd  = lds + Pdim * Edim;   // [P]    jump_dest copy, 1 KB

    const int b    = blockIdx.x;
    const int lane = threadIdx.x;     // 0..31
    const int e0   = lane * 4;

    // Zero the ring memory; stage jump_dest into LDS (single wave: DS ops
    // are in-order, no barrier needed before first use).
    const float4 z4 = make_float4(0.f, 0.f, 0.f, 0.f);
    for (int i = lane * 4; i < Pdim * Edim; i += 32 * 4)
        *(float4*)(mem + i) = z4;
#pragma unroll
    for (int i = lane; i < Pdim; i += 32)
        jd[i] = jump_dest[i];

    // Per-lane constants.
    float4 lnw4 = *(const float4*)(ln_w + e0);
    float4 lnb4 = *(const float4*)(ln_b + e0);
    float4 wg4  = *(const float4*)(Wg   + e0);
    float4 bin4 = *(const float4*)(b_in + e0);
    const float csr = ctx_strength[0];
    const float cs  = 1.0f / (1.0f + __expf(-csr));   // sigmoid
    const float bgv = bg[0];

    float hid0 = 0.f, hid1 = 0.f, hid2 = 0.f, hid3 = 0.f;
    float ptr = pointer_init[b];

    const float* __restrict__ embp = EMB + ((size_t)b * Tlen) * Edim;
    const float* __restrict__ xp   = x   + ((size_t)b * Tlen) * Idim;

    // Software pipeline: emb row for step t is loaded during step t-1.
    float4 enext = make_float4(0.f, 0.f, 0.f, 0.f);
    if (use_precomp)
        enext = *(const float4*)(embp + e0);

    for (int t = 0; t < Tlen; ++t) {
        // ---- emb = tanh(x_t @ W_in.T + b_in) for this lane's 4 elements ----
        float em0, em1, em2, em3;
        if (use_precomp) {
            em0 = enext.x; em1 = enext.y; em2 = enext.z; em3 = enext.w;
            int tn = (t + 1 < Tlen) ? (t + 1) : t;
            enext = *(const float4*)(embp + (size_t)tn * Edim + e0);   // issued early
            int tp = (t + 4 < Tlen) ? (t + 4) : t;
            __builtin_prefetch(embp + (size_t)tp * Edim + e0, 0, 3);   // warm near cache
        } else {
            const float* __restrict__ xt = xp + (size_t)t * Idim;
            float a0 = bin4.x, a1 = bin4.y, a2 = bin4.z, a3 = bin4.w;
            const float* __restrict__ w0 = Win + (size_t)(e0 + 0) * Idim;
            const float* __restrict__ w1 = Win + (size_t)(e0 + 1) * Idim;
            const float* __restrict__ w2 = Win + (size_t)(e0 + 2) * Idim;
            const float* __restrict__ w3 = Win + (size_t)(e0 + 3) * Idim;
#pragma unroll 8
            for (int i = 0; i < Idim; ++i) {
                float xv = xt[i];
                a0 = fmaf(xv, w0[i], a0);
                a1 = fmaf(xv, w1[i], a1);
                a2 = fmaf(xv, w2[i], a2);
                a3 = fmaf(xv, w3[i], a3);
            }
            em0 = FAST_TANH(a0); em1 = FAST_TANH(a1);
            em2 = FAST_TANH(a2); em3 = FAST_TANH(a3);
        }

        // ---- pointer neighborhood + softmax weights (replicated per lane) ----
        int base = (int)ptr;                       // ptr in [0,256)
        if (base < 0) base = 0;
        if (base > Pdim - 1) base = Pdim - 1;
        int   idx[5];
        float w[5];
        float wsum = 0.f;
#pragma unroll
        for (int o = 0; o < 5; ++o) {
            int id = (base + o - 2) & (Pdim - 1);  // mod 256 (two's complement AND)
            idx[o] = id;
            float d = (float)id - ptr + 128.0f;
            d -= floorf(d * (1.0f / 256.0f)) * 256.0f;   // mod 256 -> [0,256)
            d -= 128.0f;
            float s = __expf(-d * d * TEMP_INV);
            w[o] = s;
            wsum += s;
        }
        const float winv = 1.0f / wsum;
#pragma unroll
        for (int o = 0; o < 5; ++o) w[o] *= winv;

        // ---- gather: r[o] = mem[idx_o, e0..e0+3]; ctx = sum_o w_o * r[o] ----
        float4 r[5];
        float c0 = 0.f, c1 = 0.f, c2 = 0.f, c3 = 0.f;
#pragma unroll
        for (int o = 0; o < 5; ++o) {
            r[o] = *(const float4*)(mem + idx[o] * Edim + e0);
            c0 = fmaf(w[o], r[o].x, c0);
            c1 = fmaf(w[o], r[o].y, c1);
            c2 = fmaf(w[o], r[o].z, c2);
            c3 = fmaf(w[o], r[o].w, c3);
        }

        // ---- su = tanh(emb + cs*ctx + hid) ----
        float s0 = FAST_TANH(fmaf(cs, c0, em0) + hid0);
        float s1 = FAST_TANH(fmaf(cs, c1, em1) + hid1);
        float s2 = FAST_TANH(fmaf(cs, c2, em2) + hid2);
        float s3 = FAST_TANH(fmaf(cs, c3, em3) + hid3);

        // ---- LayerNorm over E=128 via wave reduction ----
        float lsum = s0 + s1 + s2 + s3;
        float lsq  = s0*s0 + s1*s1 + s2*s2 + s3*s3;
        float tsum = wave_sum32(lsum);
        float tsq  = wave_sum32(lsq);
        float mu   = tsum * (1.0f / Edim);
        float var  = tsq * (1.0f / Edim) - mu * mu;
        float rstd = rsqrtf(var + LN_EPS);
        hid0 = fmaf((s0 - mu) * rstd, lnw4.x, lnb4.x);
        hid1 = fmaf((s1 - mu) * rstd, lnw4.y, lnb4.y);
        hid2 = fmaf((s2 - mu) * rstd, lnw4.z, lnb4.z);
        hid3 = fmaf((s3 - mu) * rstd, lnw4.w, lnb4.w);

        // ---- scatter-add, reusing the gathered rows (no re-read; this wave
        //      is the only writer of this batch's rows) ----
#pragma unroll
        for (int o = 0; o < 5; ++o) {
            float4 v;
            v.x = fmaf(w[o], s0, r[o].x);
            v.y = fmaf(w[o], s1, r[o].y);
            v.z = fmaf(w[o], s2, r[o].z);
            v.w = fmaf(w[o], s3, r[o].w);
            *(float4*)(mem + idx[o] * Edim + e0) = v;
        }

        // ---- jump gate + pointer update ----
        float jl = s0*wg4.x + s1*wg4.y + s2*wg4.z + s3*wg4.w;
        jl = wave_sum32(jl) + bgv;                 // sigmoid(jl)>0.5 <=> jl>0
        float jt = jd[base];                       // LDS copy; cur == base (ptr >= 0)
        float walk = ptr + 1.0f;
        if (walk >= 256.0f) walk -= 256.0f;
        ptr = (jl > 0.0f) ? jt : walk;
    }

    // ---- logits = hid @ Wo.T + bo (share hid via LDS, reuse mem space) ----
    *(float4*)(mem + e0) = make_float4(hid0, hid1, hid2, hid3);
    __syncthreads();   // single-wave WG: lowers to S_NOP + waitcnt fence
#pragma unroll
    for (int n = lane; n < NOUT; n += 32) {
        const float* __restrict__ wr = Wo + (size_t)n * Edim;
        float acc = bo[n];
#pragma unroll 8
        for (int e = 0; e < Edim; ++e)
            acc = fmaf(mem[e], wr[e], acc);
        out[(size_t)b * NOUT + n] = acc;
    }
}

// ---------------------------------------------------------------------------
// Launcher
// ---------------------------------------------------------------------------
extern "C" void kernel_launch(void* const* d_in, const int* in_sizes, int n_in,
                              void* d_out, int out_size, void* d_ws, size_t ws_size,
                              hipStream_t stream) {
    const float* x     = (const float*)d_in[0];
    const float* pinit = (const float*)d_in[1];
    const float* Win   = (const float*)d_in[2];
    const float* b_in  = (const float*)d_in[3];
    const float* ln_w  = (const float*)d_in[4];
    const float* ln_b  = (const float*)d_in[5];
    const float* jdest = (const float*)d_in[6];
    const float* Wg    = (const float*)d_in[7];
    const float* bg    = (const float*)d_in[8];
    const float* cstr  = (const float*)d_in[9];
    const float* Wo    = (const float*)d_in[10];
    const float* bo    = (const float*)d_in[11];
    float* out = (float*)d_out;

    const size_t emb_bytes = (size_t)Bsz * Tlen * Edim * sizeof(float); // 134 MB
    const int use_precomp = (ws_size >= emb_bytes) ? 1 : 0;
    float* EMB = (float*)d_ws;

    if (use_precomp) {
        // BT/16 = 16384 M-tiles x 8 N-tiles = 131072 waves, 8 waves/block.
        emb_wmma_kernel<<<16384, 256, 0, stream>>>(x, Win, b_in, EMB);
    }

    // 128 KB ring memory + 1 KB jump table of dynamic LDS per workgroup.
    const int lds_bytes = (Pdim * Edim + Pdim) * (int)sizeof(float);
    static_cast<void>(hipFuncSetAttribute(
        reinterpret_cast<const void*>(ring_seq_kernel),
        hipFuncAttributeMaxDynamicSharedMemorySize, lds_bytes));

    ring_seq_kernel<<<Bsz, 32, lds_bytes, stream>>>(
        x, pinit, Win, b_in, ln_w, ln_b, jdest, Wg, bg, cstr, Wo, bo,
        EMB, use_precomp, out);
}